// DirectSolverNet_29480655519969
// MI455X (gfx1250) — compile-verified
//
#include <hip/hip_runtime.h>
#include <math.h>

// DirectSolverNet forward for MI455X (gfx1250, wave32).
// Stage 1: HBM-bound streaming reduction JtR = Jt @ (weights*R)  (~315 MB reads)
// Stage 2: per-batch 6x6 solve + Rodrigues + pose compose via V_WMMA_F32_16X16X4_F32.

typedef float v2f __attribute__((ext_vector_type(2)));
typedef float v4f __attribute__((ext_vector_type(4)));
typedef float v8f __attribute__((ext_vector_type(8)));

#define BATCH   64
#define NPIX    153600          // 8*120*160
#define SLICES  30              // 153600/30 = 5120 = 5 * (256*4) exact
#define TPB     256

// ---------------------------------------------------------------------------
// Kernel 1: per-(batch, slice) partial JtR.  Pure streaming: b128 loads with
// non-temporal hint (data is read exactly once; footprint 315MB > 192MB L2).
// ---------------------------------------------------------------------------
__global__ __launch_bounds__(TPB)
void jtr_partial_kernel(const float* __restrict__ Jt,
                        const float* __restrict__ W,
                        const float* __restrict__ R,
                        float* __restrict__ partial) {
  __shared__ float sred[6 * TPB];

  const int b   = blockIdx.x / SLICES;
  const int s   = blockIdx.x % SLICES;
  const int tid = threadIdx.x;

  const int    sliceLen = NPIX / SLICES;                    // 5120
  const size_t wbase    = (size_t)b * NPIX + (size_t)s * sliceLen;
  const size_t jbase    = (size_t)b * 6 * NPIX + (size_t)s * sliceLen;

  float acc[6] = {0.f, 0.f, 0.f, 0.f, 0.f, 0.f};

  // 5 exact passes of 256 threads * float4
  for (int pass = 0; pass < sliceLen / (TPB * 4); ++pass) {
    const int n = pass * TPB * 4 + tid * 4;
    v4f w4 = __builtin_nontemporal_load((const v4f*)(W + wbase + n));
    v4f r4 = __builtin_nontemporal_load((const v4f*)(R + wbase + n));
    v4f wr = w4 * r4;
#pragma unroll
    for (int r = 0; r < 6; ++r) {
      v4f j = __builtin_nontemporal_load((const v4f*)(Jt + jbase + (size_t)r * NPIX + n));
      acc[r] += j[0] * wr[0] + j[1] * wr[1] + j[2] * wr[2] + j[3] * wr[3];
    }
  }

#pragma unroll
  for (int r = 0; r < 6; ++r) sred[r * TPB + tid] = acc[r];
  __syncthreads();

  for (int off = TPB / 2; off > 0; off >>= 1) {
    if (tid < off) {
#pragma unroll
      for (int r = 0; r < 6; ++r)
        sred[r * TPB + tid] += sred[r * TPB + tid + off];
    }
    __syncthreads();
  }
  if (tid < 6) partial[((size_t)b * SLICES + s) * 6 + tid] = sred[tid * TPB];
}

// ---------------------------------------------------------------------------
// Kernel 2: one wave32 per batch.
//   lanes 0..5: sum slice partials -> JtR
//   lane 0:     H = JtJ + (trace*1e-6 + 1e-6)*I ; Gauss-Jordan solve (SPD);
//               Rodrigues dR = exp(skew(-xi[:3])); dt = -(dR @ xi[3:])
//   all lanes:  pose = [R0@dR | R0@dt + t0] via one V_WMMA_F32_16X16X4_F32
//               A = R0 (16x4, rows 0-2 / K 0-2 live), B = [dR|dt] (4x16,
//               rows 0-2 / cols 0-3 live), C[m][3] = t0[m].
// ---------------------------------------------------------------------------
__global__ __launch_bounds__(32)
void solve_pose_kernel(const float* __restrict__ JtJ,
                       const float* __restrict__ R0,
                       const float* __restrict__ t0,
                       const float* __restrict__ partial,
                       float* __restrict__ out) {
  const int b    = blockIdx.x;
  const int lane = threadIdx.x;        // 0..31, single wave

  __shared__ float jtr[6];
  __shared__ float Mtx[3][4];          // [dR | dt]

  if (lane < 6) {
    float ssum = 0.f;
    for (int i = 0; i < SLICES; ++i)
      ssum += partial[((size_t)b * SLICES + i) * 6 + lane];
    jtr[lane] = ssum;
  }
  __syncthreads();

  if (lane == 0) {
    const float* jj = JtJ + (size_t)b * 36;
    float tr = jj[0] + jj[7] + jj[14] + jj[21] + jj[28] + jj[35];
    float eps = tr * 1e-6f + 1e-6f;    // trace reg + inverse_update_pose reg

    float H[6][7];
#pragma unroll
    for (int i = 0; i < 6; ++i) {
#pragma unroll
      for (int j = 0; j < 6; ++j) H[i][j] = jj[i * 6 + j] + ((i == j) ? eps : 0.f);
      H[i][6] = jtr[i];
    }
    // Gauss-Jordan, no pivoting (SPD + regularized), fully unrolled -> registers
#pragma unroll
    for (int k = 0; k < 6; ++k) {
      float inv = 1.0f / H[k][k];
#pragma unroll
      for (int j = 0; j < 7; ++j) H[k][j] *= inv;
#pragma unroll
      for (int i = 0; i < 6; ++i) {
        if (i == k) continue;
        float f = H[i][k];
#pragma unroll
        for (int j = 0; j < 7; ++j) H[i][j] -= f * H[k][j];
      }
    }
    float xi[6];
#pragma unroll
    for (int i = 0; i < 6; ++i) xi[i] = H[i][6];

    // Rodrigues for w = -xi[0:3]
    float wx = -xi[0], wy = -xi[1], wz = -xi[2];
    float theta = sqrtf(wx * wx + wy * wy + wz * wz);
    float safe  = fmaxf(theta, 1e-12f);
    float aa    = sinf(theta) / safe;
    float cc    = (1.0f - cosf(theta)) / (safe * safe);

    float K[3][3]  = {{0.f, -wz, wy}, {wz, 0.f, -wx}, {-wy, wx, 0.f}};
    float K2[3][3];
#pragma unroll
    for (int i = 0; i < 3; ++i)
#pragma unroll
      for (int j = 0; j < 3; ++j)
        K2[i][j] = K[i][0] * K[0][j] + K[i][1] * K[1][j] + K[i][2] * K[2][j];

    float dR[3][3];
#pragma unroll
    for (int i = 0; i < 3; ++i)
#pragma unroll
      for (int j = 0; j < 3; ++j)
        dR[i][j] = ((i == j) ? 1.0f : 0.0f) + aa * K[i][j] + cc * K2[i][j];

#pragma unroll
    for (int i = 0; i < 3; ++i) {
      Mtx[i][0] = dR[i][0];
      Mtx[i][1] = dR[i][1];
      Mtx[i][2] = dR[i][2];
      Mtx[i][3] = -(dR[i][0] * xi[3] + dR[i][1] * xi[4] + dR[i][2] * xi[5]);
    }
  }
  __syncthreads();

  // Build fragments (ISA 7.12.2 layouts, wave32; EXEC is all-ones here).
  const float* r0p = R0 + (size_t)b * 9;
  const int rc  = lane & 15;           // A: row index / B: column index
  const int kHi = lane >> 4;           // 0 -> K={0,1}, 1 -> K={2,3}

  v2f afrag;                           // A[rc][2*kHi + v], R0 padded 3x3 -> 16x4
  {
    int k0 = 2 * kHi, k1 = 2 * kHi + 1;
    afrag[0] = (rc < 3 && k0 < 3) ? r0p[rc * 3 + k0] : 0.f;
    afrag[1] = (rc < 3 && k1 < 3) ? r0p[rc * 3 + k1] : 0.f;
  }
  v2f bfrag;                           // B[2*kHi + v][rc], [dR|dt] padded 3x4 -> 4x16
  {
    int k0 = 2 * kHi, k1 = 2 * kHi + 1;
    bfrag[0] = (rc < 4 && k0 < 3) ? Mtx[k0][rc] : 0.f;
    bfrag[1] = (rc < 4 && k1 < 3) ? Mtx[k1][rc] : 0.f;
  }
  v8f cfrag = {};                      // C[m][3] = t0[m]  (lane 3 holds column 3)
  if (lane == 3) {
    cfrag[0] = t0[(size_t)b * 3 + 0];
    cfrag[1] = t0[(size_t)b * 3 + 1];
    cfrag[2] = t0[(size_t)b * 3 + 2];
  }

  v8f dfrag = __builtin_amdgcn_wmma_f32_16x16x4_f32(
      /*neg_a=*/false, afrag, /*neg_b=*/false, bfrag,
      /*c_mod=*/(short)0, cfrag, /*reuse_a=*/false, /*reuse_b=*/false);

  // D[m][n] lives in VGPR m at lane n (m<8, lanes 0-15). Pose is [3,4].
  if (lane < 4) {
#pragma unroll
    for (int m = 0; m < 3; ++m)
      out[(size_t)b * 12 + m * 4 + lane] = dfrag[m];
  }
}

// ---------------------------------------------------------------------------
extern "C" void kernel_launch(void* const* d_in, const int* in_sizes, int n_in,
                              void* d_out, int out_size, void* d_ws, size_t ws_size,
                              hipStream_t stream) {
  const float* JtJ = (const float*)d_in[0];   // [64,6,6]
  const float* Jt  = (const float*)d_in[1];   // [64,6,153600]
  const float* W   = (const float*)d_in[2];   // [64,8,120,160]
  const float* R   = (const float*)d_in[3];   // [64,8,120,160]
  const float* R0  = (const float*)d_in[4];   // [64,3,3]
  const float* t0  = (const float*)d_in[5];   // [64,3,1]
  // d_in[6..10] (invD0, invD1, x0, x1, K) unused by the reference.
  float* out     = (float*)d_out;             // [64,3,4]
  float* partial = (float*)d_ws;              // [64][SLICES][6] = 46,080 B

  jtr_partial_kernel<<<BATCH * SLICES, TPB, 0, stream>>>(Jt, W, R, partial);
  solve_pose_kernel<<<BATCH, 32, 0, stream>>>(JtJ, R0, t0, partial, out);
}